// GATModel_28037546508299
// MI455X (gfx1250) — compile-verified
//
#include <hip/hip_runtime.h>

#define NN        50000
#define E_RAW     800000
#define E_TOT     (E_RAW + NN)       // edges + self loops = 850000
#define NGRAPH    2500
#define CH        64                 // HID
#define HC        128                // HEADS*HID
#define NEG_SLOPE 0.2f
#define ORD_NEG_INF 0x007FFFFFu      // ordered-uint encoding of -inf

typedef __attribute__((ext_vector_type(2))) float v2f;
typedef __attribute__((ext_vector_type(8))) float v8f;

// ---------------- ordered-uint float max helpers -------------------------
__device__ __forceinline__ unsigned f2ord(float f) {
  unsigned u = __float_as_uint(f);
  return (u & 0x80000000u) ? ~u : (u | 0x80000000u);
}
__device__ __forceinline__ float ord2f(unsigned u) {
  return (u & 0x80000000u) ? __uint_as_float(u & 0x7FFFFFFFu)
                           : __uint_as_float(~u);
}

// ---------------- fp32 WMMA GEMM: C[M,128] = A[M,K] @ B[K,128] -----------
// One wave computes a 16(M) x 128(N) strip: 8 independent v8f accumulators.
// Per k-step: load A frag once (reused across all 8 N-tiles), then issue 8
// independent V_WMMA_F32_16X16X4_F32 (no accumulator RAW chain between them).
// A frag: lane = m (lane&15), VGPR v + lane-half -> k = (lane>>4)*2 + v
// B frag: lane = n (lane&15), VGPR v + lane-half -> k = (lane>>4)*2 + v
// C frag: lane = n (lane&15), VGPR v -> m = v + (lane>>4)*8
__global__ __launch_bounds__(256) void gemm_wmma_f32(
    const float* __restrict__ A, const float* __restrict__ B,
    float* __restrict__ C, int M, int K) {
  const int lane = threadIdx.x & 31;
  const int wave = threadIdx.x >> 5;
  const int mtile = blockIdx.x * 8 + wave;
  if (mtile * 16 >= M) return;                 // wave-uniform: EXEC stays all-1
  const int m0 = mtile * 16;
  const int r    = lane & 15;
  const int half = lane >> 4;                  // 0 or 1

  const float* arow = A + (size_t)(m0 + r) * K + half * 2;
  v8f acc[8] = {};
  for (int k0 = 0; k0 < K; k0 += 4) {
    v2f a;
    a.x = arow[k0];
    a.y = arow[k0 + 1];
    const int kb = k0 + half * 2;
    const float* b0 = B + (size_t)kb * HC + r;
    const float* b1 = b0 + HC;
#pragma unroll
    for (int t = 0; t < 8; ++t) {
      v2f b;
      b.x = b0[t * 16];
      b.y = b1[t * 16];
      acc[t] = __builtin_amdgcn_wmma_f32_16x16x4_f32(false, a, false, b,
                                                     (short)0, acc[t],
                                                     false, false);
    }
  }
  float* cbase = C + (size_t)(m0 + half * 8) * HC + r;
#pragma unroll
  for (int t = 0; t < 8; ++t) {
    float* cp = cbase + t * 16;
#pragma unroll
    for (int v = 0; v < 8; ++v) cp[(size_t)v * HC] = acc[t][v];
  }
}

// ---------------- per-node attention coefficients ------------------------
__global__ __launch_bounds__(256) void node_attn(
    const float* __restrict__ h, const float* __restrict__ att_src,
    const float* __restrict__ att_dst, float* __restrict__ a_src,
    float* __restrict__ a_dst) {
  int idx = blockIdx.x * blockDim.x + threadIdx.x;   // (node, head)
  if (idx >= NN * 2) return;
  int n = idx >> 1, hd = idx & 1;
  const float* hp = h + (size_t)n * HC + hd * CH;
  const float* as = att_src + hd * CH;
  const float* ad = att_dst + hd * CH;
  float s1 = 0.f, s2 = 0.f;
#pragma unroll 8
  for (int c = 0; c < CH; ++c) { float v = hp[c]; s1 += v * as[c]; s2 += v * ad[c]; }
  a_src[idx] = s1;
  a_dst[idx] = s2;
}

// ---------------- per-layer scratch init ---------------------------------
__global__ __launch_bounds__(256) void init_layer(
    float* __restrict__ acc, unsigned* __restrict__ m, float* __restrict__ s) {
  int idx = blockIdx.x * blockDim.x + threadIdx.x;
  if (idx < NN * HC) acc[idx] = 0.f;
  if (idx < NN * 2) { m[idx] = ORD_NEG_INF; s[idx] = 0.f; }
}

// ---------------- edge pass 1: logits + segment max ----------------------
__global__ __launch_bounds__(256) void edge_max(
    const int* __restrict__ ei, const float* __restrict__ a_src,
    const float* __restrict__ a_dst, float* __restrict__ e_buf,
    unsigned* __restrict__ m_buf) {
  int idx = blockIdx.x * blockDim.x + threadIdx.x;   // (edge, head)
  if (idx >= E_TOT * 2) return;
  int e = idx >> 1, hd = idx & 1;
  int s = (e < E_RAW) ? ei[e] : (e - E_RAW);
  int d = (e < E_RAW) ? ei[E_RAW + e] : (e - E_RAW);
  float v = a_src[s * 2 + hd] + a_dst[d * 2 + hd];
  v = (v > 0.f) ? v : NEG_SLOPE * v;                 // leaky_relu
  e_buf[idx] = v;
  atomicMax(&m_buf[d * 2 + hd], f2ord(v));
}

// ---------------- edge pass 2: exp + segment sum -------------------------
__global__ __launch_bounds__(256) void edge_expsum(
    const int* __restrict__ ei, float* __restrict__ e_buf,
    const unsigned* __restrict__ m_buf, float* __restrict__ s_buf) {
  int idx = blockIdx.x * blockDim.x + threadIdx.x;
  if (idx >= E_TOT * 2) return;
  int e = idx >> 1, hd = idx & 1;
  int d = (e < E_RAW) ? ei[E_RAW + e] : (e - E_RAW);
  float m = ord2f(m_buf[d * 2 + hd]);
  float ex = __expf(e_buf[idx] - m);
  e_buf[idx] = ex;
  atomicAdd(&s_buf[d * 2 + hd], ex);
}

// ---------------- edge pass 3: weighted scatter aggregation --------------
// one wave per edge; lane covers 4 channels (float4 gather, 4 atomic adds)
__global__ __launch_bounds__(256) void edge_aggr(
    const int* __restrict__ ei, const float* __restrict__ h,
    const float* __restrict__ e_buf, const float* __restrict__ s_buf,
    float* __restrict__ acc) {
  int t = blockIdx.x * blockDim.x + threadIdx.x;
  int e = t >> 5;
  if (e >= E_TOT) return;
  int lane = t & 31;
  int s = (e < E_RAW) ? ei[e] : (e - E_RAW);
  int d = (e < E_RAW) ? ei[E_RAW + e] : (e - E_RAW);
  int hd = lane >> 4;                                // channels [lane*4, lane*4+3]
  float ex = e_buf[e * 2 + hd];
  float sv = s_buf[d * 2 + hd];
  float alpha = ex / fmaxf(sv, 1e-16f);
  const float4 hv = *(const float4*)(h + (size_t)s * HC + lane * 4);
  float* ap = acc + (size_t)d * HC + lane * 4;
  atomicAdd(ap + 0, hv.x * alpha);
  atomicAdd(ap + 1, hv.y * alpha);
  atomicAdd(ap + 2, hv.z * alpha);
  atomicAdd(ap + 3, hv.w * alpha);
}

// ---------------- head mean + bias + relu --------------------------------
__global__ __launch_bounds__(256) void node_finalize(
    const float* __restrict__ acc, const float* __restrict__ bias,
    float* __restrict__ out) {
  int idx = blockIdx.x * blockDim.x + threadIdx.x;   // (node, channel)
  if (idx >= NN * CH) return;
  int n = idx >> 6, c = idx & 63;
  float v = 0.5f * (acc[(size_t)n * HC + c] + acc[(size_t)n * HC + CH + c]) + bias[c];
  out[idx] = fmaxf(v, 0.f);
}

// ---------------- global mean pool + final linear ------------------------
__global__ __launch_bounds__(256) void pool_init(float* __restrict__ sums,
                                                 float* __restrict__ cnt) {
  int idx = blockIdx.x * blockDim.x + threadIdx.x;
  if (idx < NGRAPH * CH) sums[idx] = 0.f;
  if (idx < NGRAPH) cnt[idx] = 0.f;
}

__global__ __launch_bounds__(256) void pool_accum(
    const float* __restrict__ hn, const int* __restrict__ batch,
    float* __restrict__ sums, float* __restrict__ cnt) {
  int idx = blockIdx.x * blockDim.x + threadIdx.x;   // (node, channel)
  if (idx >= NN * CH) return;
  int n = idx >> 6, c = idx & 63;
  int g = batch[n];
  atomicAdd(&sums[(size_t)g * CH + c], hn[idx]);
  if (c == 0) atomicAdd(&cnt[g], 1.0f);
}

__global__ __launch_bounds__(256) void final_linear(
    const float* __restrict__ sums, const float* __restrict__ cnt,
    const float* __restrict__ lw, const float* __restrict__ lb,
    float* __restrict__ out) {
  int g = blockIdx.x * blockDim.x + threadIdx.x;
  if (g >= NGRAPH) return;
  float inv = 1.0f / fmaxf(cnt[g], 1.0f);
  float acc = 0.f;
#pragma unroll 8
  for (int c = 0; c < CH; ++c) acc += sums[(size_t)g * CH + c] * inv * lw[c];
  out[g] = acc + lb[0];
}

// ---------------- one GAT layer ------------------------------------------
static void run_layer(const float* in, int K, const float* W, const float* as,
                      const float* ad, const float* b, const int* ei,
                      float* h_buf, float* acc_buf, float* node_out,
                      float* a_src_b, float* a_dst_b, unsigned* m_buf,
                      float* s_buf, float* e_buf, hipStream_t stream) {
  dim3 ggrid((NN / 16 + 7) / 8);                     // 3125 M-tiles, 8 waves/block
  gemm_wmma_f32<<<ggrid, 256, 0, stream>>>(in, W, h_buf, NN, K);
  node_attn<<<(NN * 2 + 255) / 256, 256, 0, stream>>>(h_buf, as, ad, a_src_b, a_dst_b);
  init_layer<<<(NN * HC + 255) / 256, 256, 0, stream>>>(acc_buf, m_buf, s_buf);
  edge_max<<<(E_TOT * 2 + 255) / 256, 256, 0, stream>>>(ei, a_src_b, a_dst_b, e_buf, m_buf);
  edge_expsum<<<(E_TOT * 2 + 255) / 256, 256, 0, stream>>>(ei, e_buf, m_buf, s_buf);
  edge_aggr<<<(E_TOT * 32 + 255) / 256, 256, 0, stream>>>(ei, h_buf, e_buf, s_buf, acc_buf);
  node_finalize<<<(NN * CH + 255) / 256, 256, 0, stream>>>(acc_buf, b, node_out);
}

extern "C" void kernel_launch(void* const* d_in, const int* in_sizes, int n_in,
                              void* d_out, int out_size, void* d_ws, size_t ws_size,
                              hipStream_t stream) {
  // setup_inputs() order:
  // 0:x 1:edge_index 2:batch 3:W1 4:att_src1 5:att_dst1 6:b1
  // 7:W2 8:att_src2 9:att_dst2 10:b2 11:W3 12:att_src3 13:att_dst3 14:b3
  // 15:lin_w 16:lin_b
  const float* x     = (const float*)d_in[0];
  const int*   ei    = (const int*)d_in[1];
  const int*   batch = (const int*)d_in[2];
  const float* W[3]   = {(const float*)d_in[3], (const float*)d_in[7],  (const float*)d_in[11]};
  const float* asv[3] = {(const float*)d_in[4], (const float*)d_in[8],  (const float*)d_in[12]};
  const float* adv[3] = {(const float*)d_in[5], (const float*)d_in[9],  (const float*)d_in[13]};
  const float* bv[3]  = {(const float*)d_in[6], (const float*)d_in[10], (const float*)d_in[14]};
  const float* lin_w = (const float*)d_in[15];
  const float* lin_b = (const float*)d_in[16];
  float* out = (float*)d_out;

  // workspace carve-up (floats); total ~18.27M floats = ~73 MB
  float* ws        = (float*)d_ws;
  float* h_buf     = ws;                         // NN*HC   = 6,400,000
  float* acc_buf   = h_buf + (size_t)NN * HC;    // NN*HC   = 6,400,000
  float* node_buf  = acc_buf + (size_t)NN * HC;  // NN*CH   = 3,200,000
  float* a_src_b   = node_buf + (size_t)NN * CH; // NN*2    =   100,000
  float* a_dst_b   = a_src_b + NN * 2;           // NN*2
  unsigned* m_buf  = (unsigned*)(a_dst_b + NN * 2); // NN*2
  float* s_buf     = (float*)(m_buf + NN * 2);   // NN*2
  float* e_buf     = s_buf + NN * 2;             // E_TOT*2 = 1,700,000
  float* sums      = e_buf + (size_t)E_TOT * 2;  // NGRAPH*CH = 160,000
  float* cnt       = sums + (size_t)NGRAPH * CH; // NGRAPH

  run_layer(x,        128, W[0], asv[0], adv[0], bv[0], ei, h_buf, acc_buf,
            node_buf, a_src_b, a_dst_b, m_buf, s_buf, e_buf, stream);
  run_layer(node_buf,  64, W[1], asv[1], adv[1], bv[1], ei, h_buf, acc_buf,
            node_buf, a_src_b, a_dst_b, m_buf, s_buf, e_buf, stream);
  run_layer(node_buf,  64, W[2], asv[2], adv[2], bv[2], ei, h_buf, acc_buf,
            node_buf, a_src_b, a_dst_b, m_buf, s_buf, e_buf, stream);

  pool_init<<<(NGRAPH * CH + 255) / 256, 256, 0, stream>>>(sums, cnt);
  pool_accum<<<(NN * CH + 255) / 256, 256, 0, stream>>>(node_buf, batch, sums, cnt);
  final_linear<<<(NGRAPH + 255) / 256, 256, 0, stream>>>(sums, cnt, lin_w, lin_b, out);
}